// Conv_39273180955616
// MI455X (gfx1250) — compile-verified
//
#include <hip/hip_runtime.h>

typedef __attribute__((ext_vector_type(2))) float v2f;
typedef __attribute__((ext_vector_type(8))) float v8f;

#define HH 512
#define WW 512
#define PATCH_ROWS 22    // 16 + 6 halo rows
#define PATCH_COLS 24    // 16 + 6 halo cols, padded to multiple of 4 (K-dim)
#define PATCH_STRIDE 26  // LDS row stride (floats), even for b64 alignment
#define WAVES_PER_BLOCK 8
#define TILES_PER_WAVE 4

// Per kernel row i:  D(16x16) += A(16x24 image slab) x B_i(24x16 banded kernel)
// A[m,k] = Xpad[h0+m+i-3, w0+k-3];  B_i[k,n] = K[i, k-n] for 0<=k-n<=6 else 0.
__global__ __launch_bounds__(256)
void conv7x7_wmma_f32(const float* __restrict__ X,
                      const float* __restrict__ Kw,
                      const float* __restrict__ bptr,
                      float* __restrict__ out) {
    __shared__ float Bt[7][16][PATCH_COLS];   // banded kernel, transposed: Bt[i][n][k]
    __shared__ float patch[WAVES_PER_BLOCK][PATCH_ROWS][PATCH_STRIDE];

    const int tid = threadIdx.x;

    // ---- build the 7 banded B matrices once per block (7*16*24 = 2688) ----
    for (int t = tid; t < 7 * 16 * PATCH_COLS; t += 256) {
        const int i   = t / (16 * PATCH_COLS);
        const int rem = t % (16 * PATCH_COLS);
        const int n   = rem / PATCH_COLS;
        const int k   = rem % PATCH_COLS;
        const int d   = k - n;
        Bt[i][n][k] = (d >= 0 && d < 7) ? Kw[i * 7 + d] : 0.0f;
    }
    __syncthreads();

    const int wave = tid >> 5;        // wave32
    const int lane = tid & 31;
    const int m    = lane & 15;       // A: M row ; B: N col ; D: N col
    const int o    = (lane >> 4) << 1;// K sub-offset: lanes 0-15 -> K+0,1 ; 16-31 -> K+2,3
    const float bias = bptr[0] * 49.0f;

    for (int j = 0; j < TILES_PER_WAVE; ++j) {
        const int tile_id = (blockIdx.x * WAVES_PER_BLOCK + wave) * TILES_PER_WAVE + j;
        const int plane   = tile_id >> 10;           // 32x32 tiles per 512x512 plane
        const int trem    = tile_id & 1023;
        const int h0      = (trem >> 5) << 4;
        const int w0      = (trem & 31) << 4;
        const float* Xp   = X + (size_t)plane * (HH * WW);

        // Async LDS writes are NOT ordered vs DS reads: drain previous tile's
        // WMMA operand loads before overwriting the patch.
        asm volatile("s_wait_dscnt 0" ::: "memory");

        // ---- stage 22x24 halo patch: async memory->LDS, zeros at the border ----
        for (int t = lane; t < PATCH_ROWS * PATCH_COLS; t += 32) {
            const int rr = t / PATCH_COLS;
            const int cc = t % PATCH_COLS;
            const int r  = h0 + rr - 3;
            const int c  = w0 + cc - 3;
            const unsigned lds_addr = (unsigned)(uintptr_t)&patch[wave][rr][cc];
            if (r >= 0 && r < HH && c >= 0 && c < WW) {
                const float* g = Xp + r * WW + c;
                asm volatile("global_load_async_to_lds_b32 %0, %1, off"
                             :: "v"(lds_addr), "v"(g) : "memory");
            } else {
                patch[wave][rr][cc] = 0.0f;   // zero padding
            }
        }
        asm volatile("s_wait_asynccnt 0\n\ts_wait_dscnt 0" ::: "memory");
        __builtin_amdgcn_wave_barrier();

        // ---- 7 kernel rows x 6 K-steps of v_wmma_f32_16x16x4_f32 ----
        v8f acc = {};
#pragma unroll
        for (int i = 0; i < 7; ++i) {
            const float* arow = &patch[wave][m + i][0];  // X row h0+m+i-3
            const float* brow = &Bt[i][m][0];            // Bt[i][n=m][k]
#pragma unroll
            for (int kk = 0; kk < 6; ++kk) {
                const int k = kk * 4 + o;
                v2f a = *(const v2f*)(arow + k);
                v2f b = *(const v2f*)(brow + k);
                // (neg_a, A, neg_b, B, c_mod, C, reuse_a, reuse_b)
                acc = __builtin_amdgcn_wmma_f32_16x16x4_f32(
                    false, a, false, b, (short)0, acc, false, false);
            }
        }

        // ---- epilogue: + b*49 ; C/D layout: VGPR r -> M=r (lanes<16) / M=8+r ----
        float* Op = out + (size_t)plane * (HH * WW);
        const int col   = w0 + m;
        const int rbase = h0 + ((lane >> 4) << 3);
#pragma unroll
        for (int r = 0; r < 8; ++r) {
            Op[(size_t)(rbase + r) * WW + col] = acc[r] + bias;
        }
    }
}

extern "C" void kernel_launch(void* const* d_in, const int* in_sizes, int n_in,
                              void* d_out, int out_size, void* d_ws, size_t ws_size,
                              hipStream_t stream) {
    const float* X  = (const float*)d_in[0];
    const float* Kw = (const float*)d_in[1];
    const float* b  = (const float*)d_in[2];
    float* out      = (float*)d_out;

    const int planes = in_sizes[0] / (HH * WW);                       // 1024
    const int tiles  = planes * (HH / 16) * (WW / 16);                // 1,048,576
    const int blocks = tiles / (WAVES_PER_BLOCK * TILES_PER_WAVE);    // 32,768

    hipLaunchKernelGGL(conv7x7_wmma_f32, dim3(blocks), dim3(256), 0, stream,
                       X, Kw, b, out);
}